// TimedGATConv_8761733284363
// MI455X (gfx1250) — compile-verified
//
#include <hip/hip_runtime.h>
#include <hip/hip_bf16.h>
#include <cstdint>
#include <cstddef>

#define NN 100000
#define EE 1600000
#define INF_FEATS 128
#define HH 4
#define FFH 32
#define HF 128            // H*F
#define NEG_SLOPE 0.2f

typedef __attribute__((ext_vector_type(2))) float v2f;
typedef __attribute__((ext_vector_type(8))) float v8f;

// ---------------------------------------------------------------------------
// Kernel 1: init  rst = bias (broadcast), m = -inf, denom = 0
// ---------------------------------------------------------------------------
__global__ void gat_init_kernel(float* __restrict__ rst, const float* __restrict__ bias,
                                float* __restrict__ mbuf, float* __restrict__ denom) {
    int idx = blockIdx.x * blockDim.x + threadIdx.x;
    if (idx < NN * HF) {
        rst[idx] = bias[idx & (HF - 1)];
    }
    if (idx < NN * HH) {
        mbuf[idx]  = -INFINITY;
        denom[idx] = 0.0f;
    }
}

// ---------------------------------------------------------------------------
// Kernel 1b: repack W[128,128] into WMMA B-fragment register order so the GEMM
// hot loop fetches 2 K-steps of a B fragment with one global_load_b128.
//   Wr[((t*32 + lane)*64) + kk*2 + j] = W[kk*4 + (lane>>4)*2 + j][t*16 + (lane&15)]
//   (t = col tile 0..7, lane = 0..31, kk = 0..31, j = 0..1)
// ---------------------------------------------------------------------------
__global__ void gat_wrepack_kernel(const float* __restrict__ W, float* __restrict__ Wr) {
    int idx = blockIdx.x * blockDim.x + threadIdx.x;   // 0 .. 16383
    if (idx >= HF * HF) return;
    const int t    = idx >> 11;          // /2048
    const int lane = (idx >> 6) & 31;
    const int r    = idx & 63;
    const int kk   = r >> 1;
    const int j    = r & 1;
    const int row  = kk * 4 + (lane >> 4) * 2 + j;
    const int col  = t * 16 + (lane & 15);
    Wr[idx] = W[row * HF + col];
}

// ---------------------------------------------------------------------------
// Kernel 2: WMMA GEMM  feat[N,128] @ W[128,128] -> feat_src[N,128]
// One wave computes a 16-row x 128-col strip: 8 C-tiles, K-loop of 32 x (K=4),
// processed as 16 K-pairs. B fragments come from the repacked Wr via b128 loads.
// A frag (16x4 f32): lanes 0-15 row M=lane&15, K={0,1}; lanes 16-31 K={2,3}.
// C tile: VGPR r -> row (base + half*8 + r), col = lane&15.
// ---------------------------------------------------------------------------
__global__ void __launch_bounds__(128)
gat_gemm_wmma_kernel(const float* __restrict__ feat, const float* __restrict__ Wr,
                     float* __restrict__ feat_src) {
    const int wave    = threadIdx.x >> 5;          // 0..3
    const int lane    = threadIdx.x & 31;
    const int l16     = lane & 15;
    const int half    = lane >> 4;                 // 0 or 1
    const int rowBase = (blockIdx.x * 4 + wave) * 16;

    int arow = rowBase + l16;
    if (arow >= NN) arow = NN - 1;                 // clamp tail reads; stores are guarded

    v8f acc[8];
#pragma unroll
    for (int t = 0; t < 8; ++t) acc[t] = (v8f){0.f,0.f,0.f,0.f,0.f,0.f,0.f,0.f};

    const float* arowp = feat + (size_t)arow * INF_FEATS;

#pragma unroll 4
    for (int kkp = 0; kkp < 16; ++kkp) {           // K-pair: covers kk = 2*kkp, 2*kkp+1
        const int k0 = kkp * 8 + half * 2;
        v2f a0 = *(const v2f*)(arowp + k0);        // A frag for kk = 2*kkp
        v2f a1 = *(const v2f*)(arowp + k0 + 4);    // A frag for kk = 2*kkp+1

        v2f b1s[8];
#pragma unroll
        for (int t = 0; t < 8; ++t) {
            const float4 bq = *(const float4*)(Wr + ((size_t)(t * 32 + lane)) * 64 + kkp * 4);
            v2f b0; b0.x = bq.x; b0.y = bq.y;
            v2f b1; b1.x = bq.z; b1.y = bq.w;
            acc[t] = __builtin_amdgcn_wmma_f32_16x16x4_f32(
                false, a0, false, b0, (short)0, acc[t], false, false);
            b1s[t] = b1;
        }
#pragma unroll
        for (int t = 0; t < 8; ++t) {
            acc[t] = __builtin_amdgcn_wmma_f32_16x16x4_f32(
                false, a1, false, b1s[t], (short)0, acc[t], false, false);
        }
    }

#pragma unroll
    for (int t = 0; t < 8; ++t) {
#pragma unroll
        for (int r = 0; r < 8; ++r) {
            const int row = rowBase + half * 8 + r;
            if (row < NN) {
                feat_src[(size_t)row * HF + t * 16 + l16] = acc[t][r];
            }
        }
    }
}

// ---------------------------------------------------------------------------
// Kernel 3: per-node attention halves  el[n,h], er[n,h]
// ---------------------------------------------------------------------------
__global__ void gat_eler_kernel(const float* __restrict__ feat_src,
                                const float* __restrict__ attn_l,
                                const float* __restrict__ attn_r,
                                float* __restrict__ el, float* __restrict__ er) {
    int idx = blockIdx.x * blockDim.x + threadIdx.x;   // n*H + h
    if (idx >= NN * HH) return;
    const int n = idx >> 2;
    const int h = idx & 3;
    const float* fp = feat_src + (size_t)n * HF + h * FFH;
    const float* al = attn_l + h * FFH;
    const float* ar = attn_r + h * FFH;
    float sl = 0.f, sr = 0.f;
#pragma unroll
    for (int f = 0; f < FFH; f += 4) {
        float4 v = *(const float4*)(fp + f);
        float4 a = *(const float4*)(al + f);
        float4 b = *(const float4*)(ar + f);
        sl += v.x * a.x + v.y * a.y + v.z * a.z + v.w * a.w;
        sr += v.x * b.x + v.y * b.y + v.z * b.z + v.w * b.w;
    }
    el[idx] = sl;
    er[idx] = sr;
}

// ---------------------------------------------------------------------------
// float atomic-max via signed/unsigned int atomics (init must be -inf)
// ---------------------------------------------------------------------------
__device__ __forceinline__ void atomicMaxFloat(float* addr, float v) {
    if (v >= 0.0f) atomicMax((int*)addr, __float_as_int(v));
    else           atomicMin((unsigned int*)addr, __float_as_uint(v));
}

__device__ __forceinline__ float leaky(float x) {
    return x > 0.0f ? x : NEG_SLOPE * x;
}

// ---------------------------------------------------------------------------
// Kernel 4: per-edge scores -> segment max over dst
// ---------------------------------------------------------------------------
__global__ void gat_edge_max_kernel(const int* __restrict__ src, const int* __restrict__ dst,
                                    const float* __restrict__ el, const float* __restrict__ er,
                                    float* __restrict__ mbuf) {
    int e = blockIdx.x * blockDim.x + threadIdx.x;
    if (e >= EE) return;
    const int s = src[e], d = dst[e];
    float4 l = *(const float4*)(el + (size_t)s * HH);
    float4 r = *(const float4*)(er + (size_t)d * HH);
    atomicMaxFloat(mbuf + (size_t)d * HH + 0, leaky(l.x + r.x));
    atomicMaxFloat(mbuf + (size_t)d * HH + 1, leaky(l.y + r.y));
    atomicMaxFloat(mbuf + (size_t)d * HH + 2, leaky(l.z + r.z));
    atomicMaxFloat(mbuf + (size_t)d * HH + 3, leaky(l.w + r.w));
}

// ---------------------------------------------------------------------------
// Kernel 5: ex = exp(e - m[dst]); denom[dst] += ex; stash ex per edge
// ---------------------------------------------------------------------------
__global__ void gat_edge_expsum_kernel(const int* __restrict__ src, const int* __restrict__ dst,
                                       const float* __restrict__ el, const float* __restrict__ er,
                                       const float* __restrict__ mbuf,
                                       float* __restrict__ exbuf, float* __restrict__ denom) {
    int e = blockIdx.x * blockDim.x + threadIdx.x;
    if (e >= EE) return;
    const int s = src[e], d = dst[e];
    float4 l = *(const float4*)(el + (size_t)s * HH);
    float4 r = *(const float4*)(er + (size_t)d * HH);
    float4 m = *(const float4*)(mbuf + (size_t)d * HH);
    float4 ex;
    ex.x = __expf(leaky(l.x + r.x) - m.x);
    ex.y = __expf(leaky(l.y + r.y) - m.y);
    ex.z = __expf(leaky(l.z + r.z) - m.z);
    ex.w = __expf(leaky(l.w + r.w) - m.w);
    *(float4*)(exbuf + (size_t)e * HH) = ex;
    float* dn = denom + (size_t)d * HH;
    unsafeAtomicAdd(dn + 0, ex.x);
    unsafeAtomicAdd(dn + 1, ex.y);
    unsafeAtomicAdd(dn + 2, ex.z);
    unsafeAtomicAdd(dn + 3, ex.w);
}

// ---------------------------------------------------------------------------
// Kernel 6: weighted scatter-add. One wave32 per edge; lane owns 4 features.
// feat_src (51 MB) and rst (51 MB) both fit in the 192 MB L2 -> on-chip traffic.
// ---------------------------------------------------------------------------
__global__ void __launch_bounds__(256)
gat_edge_aggregate_kernel(const int* __restrict__ src, const int* __restrict__ dst,
                          const float* __restrict__ exbuf, const float* __restrict__ denom,
                          const float* __restrict__ feat_src, float* __restrict__ rst) {
    const int wid  = blockIdx.x * (blockDim.x >> 5) + (threadIdx.x >> 5);  // edge id
    if (wid >= EE) return;
    const int lane = threadIdx.x & 31;
    const int s = src[wid];
    const int d = dst[wid];
    const int h = lane >> 3;                         // (lane*4)/32
    const float dn = fmaxf(denom[(size_t)d * HH + h], 1e-9f);
    const float alpha = exbuf[(size_t)wid * HH + h] / dn;

    float4 v = *(const float4*)(feat_src + (size_t)s * HF + lane * 4);
    float* rp = rst + (size_t)d * HF + lane * 4;
    unsafeAtomicAdd(rp + 0, v.x * alpha);
    unsafeAtomicAdd(rp + 1, v.y * alpha);
    unsafeAtomicAdd(rp + 2, v.z * alpha);
    unsafeAtomicAdd(rp + 3, v.w * alpha);
}

// ---------------------------------------------------------------------------
extern "C" void kernel_launch(void* const* d_in, const int* in_sizes, int n_in,
                              void* d_out, int out_size, void* d_ws, size_t ws_size,
                              hipStream_t stream) {
    const float* feat   = (const float*)d_in[0];
    const float* W      = (const float*)d_in[1];
    const float* attn_l = (const float*)d_in[2];
    const float* attn_r = (const float*)d_in[3];
    const float* bias   = (const float*)d_in[4];
    const int*   src    = (const int*)d_in[5];
    const int*   dst    = (const int*)d_in[6];
    float* rst = (float*)d_out;

    // workspace carve-up (floats): feat_src | el | er | m | denom | ex | Wr  (~83.3 MB)
    float* feat_src = (float*)d_ws;
    float* el    = feat_src + (size_t)NN * HF;
    float* er    = el    + (size_t)NN * HH;
    float* mbuf  = er    + (size_t)NN * HH;
    float* denom = mbuf  + (size_t)NN * HH;
    float* exbuf = denom + (size_t)NN * HH;
    float* Wr    = exbuf + (size_t)EE * HH;

    // 1) init outputs + softmax accumulators
    gat_init_kernel<<<(NN * HF + 255) / 256, 256, 0, stream>>>(rst, bias, mbuf, denom);

    // 1b) repack W into B-fragment order (64 KB, one-time)
    gat_wrepack_kernel<<<(HF * HF + 255) / 256, 256, 0, stream>>>(W, Wr);

    // 2) WMMA projection GEMM (4 waves/block, 64 rows/block)
    gat_gemm_wmma_kernel<<<(NN + 63) / 64, 128, 0, stream>>>(feat, Wr, feat_src);

    // 3) per-node attention dots
    gat_eler_kernel<<<(NN * HH + 255) / 256, 256, 0, stream>>>(feat_src, attn_l, attn_r, el, er);

    // 4) segment max
    gat_edge_max_kernel<<<(EE + 255) / 256, 256, 0, stream>>>(src, dst, el, er, mbuf);

    // 5) exp + segment sum
    gat_edge_expsum_kernel<<<(EE + 255) / 256, 256, 0, stream>>>(src, dst, el, er, mbuf,
                                                                 exbuf, denom);

    // 6) normalized scatter-add aggregation (1 wave per edge)
    gat_edge_aggregate_kernel<<<(EE + 7) / 8, 256, 0, stream>>>(src, dst, exbuf, denom,
                                                                feat_src, rst);
}